// RNN_68135361184097
// MI455X (gfx1250) — compile-verified
//
#include <hip/hip_runtime.h>

typedef float v2f __attribute__((ext_vector_type(2)));
typedef float v8f __attribute__((ext_vector_type(8)));

#define WAVES 8
#define TMAX 30
#define TRS 18   // padded LDS row stride (conflict-free C/D->A transpose)

#if __has_builtin(__builtin_amdgcn_tanhf)
#define HW_TANH(x) __builtin_amdgcn_tanhf(x)
#else
#define HW_TANH(x) tanhf(x)
#endif

__global__ __launch_bounds__(WAVES * 32) void rnn_elman_wmma(
    const float* __restrict__ x,      // [B,32]
    const int*   __restrict__ Tp,     // scalar T (device)
    const float* __restrict__ W_ih,   // [16,32]
    const float* __restrict__ W_hh,   // [16,16]
    const float* __restrict__ b_ih,   // [16]
    const float* __restrict__ b_hh,   // [16]
    const float* __restrict__ W_fc,   // [1,16]
    const float* __restrict__ b_fc,   // [1]
    float* __restrict__ out,          // [B,T]
    int batch)
{
    __shared__ float s_tr[WAVES][16 * TRS];   // per-wave transpose tile
    __shared__ float s_out[WAVES][16 * TMAX]; // per-wave staged outputs [row][t]

    const int T    = *Tp;                 // 30
    const int lane = threadIdx.x & 31;
    const int wv   = threadIdx.x >> 5;
    const int lo   = lane & 15;           // M row (A) / N col (B,C,D)
    const int hi   = lane >> 4;           // lane-half: K-pair select
    const long tile = (long)blockIdx.x * WAVES + wv;
    const long row0 = tile * 16;
    if (row0 >= batch) return;

    float* tr  = s_tr[wv];
    float* ost = s_out[wv];

    // ---- W_hh^T in B layout: B[k][j] = W_hh[j][k]; lane column j = lo,
    //      slice s covers K=4s..4s+3, this lane-half holds K=4s+2hi, +1.
    v2f wb0 = *(const v2f*)(W_hh + lo * 16 +  0 + 2 * hi);
    v2f wb1 = *(const v2f*)(W_hh + lo * 16 +  4 + 2 * hi);
    v2f wb2 = *(const v2f*)(W_hh + lo * 16 +  8 + 2 * hi);
    v2f wb3 = *(const v2f*)(W_hh + lo * 16 + 12 + 2 * hi);

    // ---- W_fc pairs aligned with the A-layout K positions this lane holds
    v2f wf0 = *(const v2f*)(W_fc +  0 + 2 * hi);
    v2f wf1 = *(const v2f*)(W_fc +  4 + 2 * hi);
    v2f wf2 = *(const v2f*)(W_fc +  8 + 2 * hi);
    v2f wf3 = *(const v2f*)(W_fc + 12 + 2 * hi);

    const float bsum = b_ih[lo] + b_hh[lo];  // C/D element (m,n): n = lo
    const float bfc  = b_fc[0];

    // ---- xproj = x @ W_ih^T + (b_ih+b_hh): 8 chained f32 WMMAs, K=32
    v8f xp;
#pragma unroll
    for (int r = 0; r < 8; ++r) xp[r] = bsum;
#pragma unroll
    for (int s = 0; s < 8; ++s) {
        // A slice: row m = lo, K = 4s+2hi, 4s+2hi+1 (consecutive -> float2)
        v2f ax = *(const v2f*)(x + (row0 + lo) * 32 + 4 * s + 2 * hi);
        // B slice: col n = lo of W_ih^T -> row n of W_ih
        v2f bx = *(const v2f*)(W_ih + lo * 32 + 4 * s + 2 * hi);
        xp = __builtin_amdgcn_wmma_f32_16x16x4_f32(false, ax, false, bx,
                                                   (short)0, xp, false, false);
    }

    // ---- recurrence: h_{t+1} = tanh(xproj + h_t @ W_hh^T), h_0 = 0
    v8f acc = xp;
    for (int t = 0; t < T; ++t) {
        // hardware V_TANH_F32 (TRANS32, co-executes with VALU)
        v8f h;
#pragma unroll
        for (int r = 0; r < 8; ++r) h[r] = HW_TANH(acc[r]);

        // C/D layout -> LDS (row m = r + 8*hi, col n = lo), stride-18 pad
#pragma unroll
        for (int r = 0; r < 8; ++r) tr[(r + 8 * hi) * TRS + lo] = h[r];

        // LDS -> A layout slices (row m = lo, K pairs per lane-half)
        v2f a0 = *(const v2f*)(tr + lo * TRS +  0 + 2 * hi);
        v2f a1 = *(const v2f*)(tr + lo * TRS +  4 + 2 * hi);
        v2f a2 = *(const v2f*)(tr + lo * TRS +  8 + 2 * hi);
        v2f a3 = *(const v2f*)(tr + lo * TRS + 12 + 2 * hi);

        // FC dot for this timestep: half-dot per lane, combine halves
        float p = a0.x * wf0.x + a0.y * wf0.y + a1.x * wf1.x + a1.y * wf1.y
                + a2.x * wf2.x + a2.y * wf2.y + a3.x * wf3.x + a3.y * wf3.y;
        p += __shfl_xor(p, 16, 32);
        if (lane < 16) ost[lo * T + t] = p + bfc;

        if (t + 1 < T) {
            acc = __builtin_amdgcn_wmma_f32_16x16x4_f32(false, a0, false, wb0,
                                                        (short)0, xp,  false, false);
            acc = __builtin_amdgcn_wmma_f32_16x16x4_f32(false, a1, false, wb1,
                                                        (short)0, acc, false, false);
            acc = __builtin_amdgcn_wmma_f32_16x16x4_f32(false, a2, false, wb2,
                                                        (short)0, acc, false, false);
            acc = __builtin_amdgcn_wmma_f32_16x16x4_f32(false, a3, false, wb3,
                                                        (short)0, acc, false, false);
        }
    }

    // ---- coalesced writeout: tile's out rows are contiguous [16*T] floats
    const int total = 16 * T;
    float* obase = out + row0 * T;
    for (int i = lane; i < total; i += 32) obase[i] = ost[i];
}

extern "C" void kernel_launch(void* const* d_in, const int* in_sizes, int n_in,
                              void* d_out, int out_size, void* d_ws, size_t ws_size,
                              hipStream_t stream) {
    const float* x    = (const float*)d_in[0];
    const int*   Tp   = (const int*)  d_in[1];
    const float* W_ih = (const float*)d_in[2];
    const float* W_hh = (const float*)d_in[3];
    const float* b_ih = (const float*)d_in[4];
    const float* b_hh = (const float*)d_in[5];
    const float* W_fc = (const float*)d_in[6];
    const float* b_fc = (const float*)d_in[7];
    float* out = (float*)d_out;

    const int batch  = in_sizes[0] / 32;
    const int tiles  = (batch + 15) / 16;
    const int blocks = (tiles + WAVES - 1) / WAVES;

    rnn_elman_wmma<<<blocks, WAVES * 32, 0, stream>>>(
        x, Tp, W_ih, W_hh, b_ih, b_hh, W_fc, b_fc, out, batch);
}